// CLIP_MINN_88210038326221
// MI455X (gfx1250) — compile-verified
//
#include <hip/hip_runtime.h>
#include <hip/hip_bf16.h>
#include <math.h>

typedef __attribute__((ext_vector_type(2))) float v2f;
typedef __attribute__((ext_vector_type(8))) float v8f;

#define QN 100      // quadrature order (NB_STEPS)
#define NSTEP 101   // QN+1 quadrature points
#define D 100       // hidden dim
#define DP 112      // D padded to 7*16
#define NT 7        // N tiles of 16
#define KC 25       // K chunks of 4 (K=100)
#define ASTR 116    // A-buffer row stride (bank-conflict padding)
#define WAVES 4
#define MPW 2       // steps (M-tiles) per wave per group
#define SPG (WAVES*MPW)  // 8 steps per group
#define NGROUP 13   // ceil(101/8) -> covers s=0..103, tail weighted 0

// ---------------- constants: Clenshaw-Curtis weights + steps ----------------
__global__ void init_const_kernel(float* wsf) {
    int s = threadIdx.x;
    if (s >= NSTEP) return;
    const double pi = 3.14159265358979323846;
    wsf[112 + s] = (float)cos((double)s * pi / (double)QN);
    double sum = 0.0;
    for (int a = 0; a <= QN; a += 2) {          // odd-a quadrature weights are 0
        double Wq = (a == 0) ? 1.0 : 2.0 / (1.0 - (double)a * (double)a);
        double l = cos((double)(a * s) * pi / (double)QN);
        if (s == 0)  l = 0.5;
        if (s == QN) l *= 0.5;
        l *= 2.0 / (double)QN;
        sum += l * Wq;
    }
    wsf[8 + s] = (float)sum;
}

// ---------------- h-MLP with zero input: 2 constants per k ------------------
__global__ __launch_bounds__(128) void oh_kernel(
    const float* __restrict__ hW1, const float* __restrict__ hb0,
    const float* __restrict__ hb1, const float* __restrict__ hW2,
    const float* __restrict__ hb2, const float* __restrict__ hW3,
    const float* __restrict__ hb3, float* __restrict__ wsf) {
    __shared__ float z0[D], z1[D], z2[D];
    const int k = blockIdx.x;
    const int tid = threadIdx.x;
    if (tid < D) z0[tid] = fmaxf(hb0[k*D + tid], 0.f);   // relu(0 @ W.T + b)
    __syncthreads();
    if (tid < D) {
        float s = hb1[k*D + tid];
        for (int i = 0; i < D; ++i) s += hW1[k*D*D + tid*D + i] * z0[i];
        z1[tid] = fmaxf(s, 0.f);
    }
    __syncthreads();
    if (tid < D) {
        float s = hb2[k*D + tid];
        for (int i = 0; i < D; ++i) s += hW2[k*D*D + tid*D + i] * z1[i];
        z2[tid] = fmaxf(s, 0.f);
    }
    __syncthreads();
    if (tid < 2) {
        float s = hb3[k*2 + tid];
        for (int i = 0; i < D; ++i) s += hW3[k*2*D + tid*D + i] * z2[i];
        wsf[tid*3 + k] = s;    // ws[0..2]=oh0[k], ws[3..5]=oh1[k]
    }
}

// ---- layer-1 partial pass over tiles [T0, T0+TN): A built inline (layer-0) --
template<int T0, int TN>
__device__ __forceinline__ void l1_pass(
    const float* __restrict__ sBp, const float* __restrict__ s_w0,
    const float* __restrict__ s_b0, const float* __restrict__ s_b1,
    float xtA, float xtB, int ln, int h,
    float* __restrict__ myA0, float* __restrict__ myA1) {
    v8f cA[TN], cB[TN];
    #pragma unroll
    for (int t = 0; t < TN; ++t) {
        float bb = s_b1[(T0 + t) * 16 + ln];
        v8f cv = {bb, bb, bb, bb, bb, bb, bb, bb};
        cA[t] = cv; cB[t] = cv;
    }
    const int lane2 = (h * 16 + ln) * 2;
    for (int kc = 0; kc < KC; ++kc) {
        const int j0 = kc * 4 + 2 * h;
        const float w00 = s_w0[j0], w01 = s_w0[j0 + 1];
        const float bb0 = s_b0[j0], bb1 = s_b0[j0 + 1];
        v2f aA, aB;
        aA.x = fmaxf(xtA * w00 + bb0, 0.f);
        aA.y = fmaxf(xtA * w01 + bb1, 0.f);
        aB.x = fmaxf(xtB * w00 + bb0, 0.f);
        aB.y = fmaxf(xtB * w01 + bb1, 0.f);
        #pragma unroll
        for (int t = 0; t < TN; ++t) {
            v2f bf = *(const v2f*)&sBp[(kc * NT + T0 + t) * 64 + lane2];
            cA[t] = __builtin_amdgcn_wmma_f32_16x16x4_f32(
                false, aA, false, bf, (short)0, cA[t], false, false);
            cB[t] = __builtin_amdgcn_wmma_f32_16x16x4_f32(
                false, aB, false, bf, (short)0, cB[t], false, false);
        }
    }
    #pragma unroll
    for (int t = 0; t < TN; ++t) {
        #pragma unroll
        for (int v = 0; v < 8; ++v) {
            const int row = v + 8 * h, col = (T0 + t) * 16 + ln;
            myA0[row * ASTR + col] = fmaxf(cA[t][v], 0.f);
            myA1[row * ASTR + col] = fmaxf(cB[t][v], 0.f);
        }
    }
}

// ---- layer-2 partial pass over tiles [T0, T0+TN): A from LDS, dot epilogue --
template<int T0, int TN>
__device__ __forceinline__ void l2_pass(
    const float* __restrict__ sBp, const float* __restrict__ s_b2,
    const float* __restrict__ s_w3, int ln, int h,
    const float* __restrict__ myA0, const float* __restrict__ myA1,
    float* __restrict__ pA, float* __restrict__ pB) {
    v8f cA[TN], cB[TN];
    #pragma unroll
    for (int t = 0; t < TN; ++t) {
        float bb = s_b2[(T0 + t) * 16 + ln];
        v8f cv = {bb, bb, bb, bb, bb, bb, bb, bb};
        cA[t] = cv; cB[t] = cv;
    }
    const int lane2 = (h * 16 + ln) * 2;
    for (int kc = 0; kc < KC; ++kc) {
        const int j0 = kc * 4 + 2 * h;
        v2f aA = *(const v2f*)&myA0[ln * ASTR + j0];   // contiguous pair -> b64
        v2f aB = *(const v2f*)&myA1[ln * ASTR + j0];
        #pragma unroll
        for (int t = 0; t < TN; ++t) {
            v2f bf = *(const v2f*)&sBp[(kc * NT + T0 + t) * 64 + lane2];
            cA[t] = __builtin_amdgcn_wmma_f32_16x16x4_f32(
                false, aA, false, bf, (short)0, cA[t], false, false);
            cB[t] = __builtin_amdgcn_wmma_f32_16x16x4_f32(
                false, aB, false, bf, (short)0, cB[t], false, false);
        }
    }
    #pragma unroll
    for (int t = 0; t < TN; ++t) {
        const float w3v = s_w3[(T0 + t) * 16 + ln];    // 0 in padded columns
        #pragma unroll
        for (int v = 0; v < 8; ++v) {
            pA[v] += fmaxf(cA[t][v], 0.f) * w3v;
            pB[v] += fmaxf(cB[t][v], 0.f) * w3v;
        }
    }
}

// ---------------- heavy stage: one mono(k, x, h=0) evaluation ---------------
__global__ __launch_bounds__(128) void stage_kernel(
    const float* __restrict__ xin,
    const float* __restrict__ iW0, const float* __restrict__ ib0,
    const float* __restrict__ iW1, const float* __restrict__ ib1,
    const float* __restrict__ iW2, const float* __restrict__ ib2,
    const float* __restrict__ iW3, const float* __restrict__ ib3,
    const float* __restrict__ wsf, int k, float* __restrict__ out) {

    // B matrices packed in WMMA-fragment order: one ds_load_b64 per fragment.
    __shared__ float sB1p[D * DP];
    __shared__ float sB2p[D * DP];
    __shared__ float sA[WAVES * MPW * 16 * ASTR]; // per-wave activation tiles
    __shared__ float s_w0[DP], s_b0[DP], s_b1[DP], s_b2[DP], s_w3[DP];
    __shared__ float s_x[16], s_zacc[16];
    __shared__ float s_ccw[NSTEP + 3], s_st[NSTEP + 3];

    const int tid  = threadIdx.x;
    const int lane = tid & 31;
    const int wave = tid >> 5;
    const int ln   = lane & 15;   // M (or N) index within tile
    const int h    = lane >> 4;   // half-wave selects K pair per ISA layout
    const int n0   = blockIdx.x * 16;

    const float* W1 = iW1 + k * D * D;
    const float* W2 = iW2 + k * D * D;
    for (int idx = tid; idx < D * DP; idx += 128) {
        // element (i = K index, j = N index); fragment-order packed destination
        const int i = idx / DP, j = idx % DP;
        const int kc = i >> 2, rem = i & 3, he = rem >> 1, c = rem & 1;
        const int t = j >> 4, lne = j & 15;
        const int dst = (kc * NT + t) * 64 + (he * 16 + lne) * 2 + c;
        const float v1 = (j < D) ? W1[j * D + i] : 0.f;  // B[i][j] = W[j][i]
        const float v2 = (j < D) ? W2[j * D + i] : 0.f;
        sB1p[dst] = v1;
        sB2p[dst] = v2;
    }
    for (int j = tid; j < DP; j += 128) {
        s_w0[j] = (j < D) ? iW0[k * 300 + j * 3] : 0.f;  // only x-column; h==0
        s_b0[j] = (j < D) ? ib0[k * D + j] : 0.f;
        s_b1[j] = (j < D) ? ib1[k * D + j] : 0.f;
        s_b2[j] = (j < D) ? ib2[k * D + j] : 0.f;
        s_w3[j] = (j < D) ? iW3[k * D + j] : 0.f;
    }
    for (int s = tid; s < NSTEP; s += 128) {
        s_ccw[s] = wsf[8 + s];
        s_st[s]  = wsf[112 + s];
    }
    if (tid < 16) { s_x[tid] = xin[n0 + tid]; s_zacc[tid] = 0.f; }
    __syncthreads();

    const float b3 = ib3[k];
    const float xm = s_x[ln];

    float accm[8];
    #pragma unroll
    for (int v = 0; v < 8; ++v) accm[v] = 0.f;

    float* myA0 = &sA[(wave * MPW + 0) * 16 * ASTR];
    float* myA1 = &sA[(wave * MPW + 1) * 16 * ASTR];

    for (int g = 0; g < NGROUP; ++g) {
        const int stepA = g * SPG + wave * MPW;
        const int stepB = stepA + 1;
        const int siA = (stepA <= QN) ? stepA : QN;   // clamp; tail gets weight 0
        const int siB = (stepB <= QN) ? stepB : QN;
        const float tA = 0.5f * (s_st[siA] + 1.f);
        const float tB = 0.5f * (s_st[siB] + 1.f);
        const float wA = (stepA <= QN) ? s_ccw[siA] : 0.f;
        const float wB = (stepB <= QN) ? s_ccw[siB] : 0.f;
        const float xtA = xm * tA, xtB = xm * tB;

        // layer 1 (two tile-halves to keep <=256 VGPRs live)
        l1_pass<0, 4>(sB1p, s_w0, s_b0, s_b1, xtA, xtB, ln, h, myA0, myA1);
        l1_pass<4, 3>(sB1p, s_w0, s_b0, s_b1, xtA, xtB, ln, h, myA0, myA1);

        // layer 2 + inline layer-3 partial dot (DS ops in-order per wave)
        float pA[8], pB[8];
        #pragma unroll
        for (int v = 0; v < 8; ++v) { pA[v] = 0.f; pB[v] = 0.f; }
        l2_pass<0, 4>(sB2p, s_b2, s_w3, ln, h, myA0, myA1, pA, pB);
        l2_pass<4, 3>(sB2p, s_b2, s_w3, ln, h, myA0, myA1, pA, pB);

        #pragma unroll
        for (int off = 1; off < 16; off <<= 1) {  // reduce across the 16 N-lanes
            #pragma unroll
            for (int v = 0; v < 8; ++v) {
                pA[v] += __shfl_xor(pA[v], off, 32);
                pB[v] += __shfl_xor(pB[v], off, 32);
            }
        }
        #pragma unroll
        for (int v = 0; v < 8; ++v) {
            const float dzA = pA[v] + b3;
            const float vA  = (dzA > 0.f) ? (dzA + 1.f) : expf(dzA); // elu+1
            const float dzB = pB[v] + b3;
            const float vB  = (dzB > 0.f) ? (dzB + 1.f) : expf(dzB);
            accm[v] += wA * vA + wB * vB;
        }
    }
    if (ln == 0) {                                // rows m = v + 8*h
        #pragma unroll
        for (int v = 0; v < 8; ++v) atomicAdd(&s_zacc[v + 8 * h], accm[v]);
    }
    __syncthreads();
    if (tid < 16) {
        const float z = s_zacc[tid] * s_x[tid] * 0.5f;
        out[n0 + tid] = expf(wsf[3 + k]) * z + wsf[k];
    }
}

__global__ void copy_kernel(const float* __restrict__ src, float* __restrict__ dst, int n) {
    int i = blockIdx.x * blockDim.x + threadIdx.x;
    if (i < n) dst[i] = src[i];
}

extern "C" void kernel_launch(void* const* d_in, const int* in_sizes, int n_in,
                              void* d_out, int out_size, void* d_ws, size_t ws_size,
                              hipStream_t stream) {
    const float* logits = (const float*)d_in[0];
    // d_in[1] = nn_id (unused by the computation)
    const float* iW0 = (const float*)d_in[2];
    const float* ib0 = (const float*)d_in[3];
    const float* iW1 = (const float*)d_in[4];
    const float* ib1 = (const float*)d_in[5];
    const float* iW2 = (const float*)d_in[6];
    const float* ib2 = (const float*)d_in[7];
    const float* iW3 = (const float*)d_in[8];
    const float* ib3 = (const float*)d_in[9];
    const float* hb0 = (const float*)d_in[11];
    const float* hW1 = (const float*)d_in[12];
    const float* hb1 = (const float*)d_in[13];
    const float* hW2 = (const float*)d_in[14];
    const float* hb2 = (const float*)d_in[15];
    const float* hW3 = (const float*)d_in[16];
    const float* hb3 = (const float*)d_in[17];

    const int N = in_sizes[0];
    float* wsf = (float*)d_ws;        // [0..5] oh, [8..108] cc_w, [112..212] steps
    float* x2  = wsf + 256;           // intermediate mono(0) output, N floats
    float* out = (float*)d_out;       // [y1 | y2 | logits]

    init_const_kernel<<<1, 128, 0, stream>>>(wsf);
    oh_kernel<<<3, 128, 0, stream>>>(hW1, hb0, hb1, hW2, hb2, hW3, hb3, wsf);

    const int nblk = N / 16;
    stage_kernel<<<nblk, 128, 0, stream>>>(logits, iW0, ib0, iW1, ib1, iW2, ib2,
                                           iW3, ib3, wsf, 0, x2);      // x2 = mono(0,x1)
    stage_kernel<<<nblk, 128, 0, stream>>>(logits, iW0, ib0, iW1, ib1, iW2, ib2,
                                           iW3, ib3, wsf, 1, out);     // y1 = mono(1,x1)
    stage_kernel<<<nblk, 128, 0, stream>>>(x2, iW0, ib0, iW1, ib1, iW2, ib2,
                                           iW3, ib3, wsf, 2, out + N); // y2 = mono(2,x2)
    copy_kernel<<<(N + 255) / 256, 256, 0, stream>>>(logits, out + 2 * N, N);
}